// HypAgg_22866405884080
// MI455X (gfx1250) — compile-verified
//
#include <hip/hip_runtime.h>
#include <hip/hip_bf16.h>

typedef float v2f __attribute__((ext_vector_type(2)));
typedef float v8f __attribute__((ext_vector_type(8)));

#define MIN_NORM  1e-15f
#define ATANH_EPS 1e-5f
#define PROJ_EPS  0.004f

// fast single-instruction hardware ops (v_rcp_f32 / v_sqrt_f32, ~1 ulp)
__device__ __forceinline__ float frcp(float x)  { return __builtin_amdgcn_rcpf(x); }
__device__ __forceinline__ float fsqrt(float x) { return __builtin_amdgcn_sqrtf(x); }

__device__ __forceinline__ float artanh_f(float z) {
  // z in [0, 1-ATANH_EPS]
  return 0.5f * __logf((1.f + z) * frcp(1.f - z));
}

#define WMMA_F32(a, b, c) \
  __builtin_amdgcn_wmma_f32_16x16x4_f32(false, (a), false, (b), (short)0, (c), false, false)

// ---------------------------------------------------------------------------
// Kernel 1: per-row stats.  x2[i]=|x_i|^2, f1[i]=x_t.w1, f2[i]=x_t.w2 where
// x_t = logmap0(x) = x * artanh(|x|)/|x|.   One wave per row (d=64 -> 2/lane).
// ---------------------------------------------------------------------------
__global__ __launch_bounds__(256) void hyp_stats(const float* __restrict__ X,
                                                 const float* __restrict__ attw,
                                                 float* __restrict__ x2o,
                                                 float* __restrict__ f1o,
                                                 float* __restrict__ f2o) {
  const int lane = threadIdx.x & 31;
  const int wave = threadIdx.x >> 5;
  const int row  = blockIdx.x * 8 + wave;

  const float a0 = X[row * 64 + lane];
  const float a1 = X[row * 64 + lane + 32];
  float s2 = a0 * a0 + a1 * a1;
  float r1 = a0 * attw[lane]      + a1 * attw[lane + 32];
  float r2 = a0 * attw[64 + lane] + a1 * attw[96 + lane];
#pragma unroll
  for (int m = 16; m >= 1; m >>= 1) {
    s2 += __shfl_xor(s2, m, 32);
    r1 += __shfl_xor(r1, m, 32);
    r2 += __shfl_xor(r2, m, 32);
  }
  if (lane == 0) {
    const float pn  = fmaxf(fsqrt(s2), MIN_NORM);
    const float fac = artanh_f(fminf(pn, 1.f - ATANH_EPS)) * frcp(pn);
    x2o[row] = s2;
    f1o[row] = fac * r1;
    f2o[row] = fac * r2;
  }
}

// ---------------------------------------------------------------------------
// Kernel 2: fused pairwise pipeline.  One wave owns a 16-row strip (i0..i0+15)
// and one of 4 j-slices.  Per 16x16 j-tile:
//   (a) G tile = X_strip * X_tile^T via 16x V_WMMA_F32_16X16X4_F32 (K=64)
//   (b) in-register elementwise: alpha,beta,sn,artanh,sigmoid -> M=w*beta,
//       row-accumulate sA += w*alpha
//   (c) M tile -> LDS (pad 17) -> A-layout frags -> GEMM2 accumulate
//       sup(16x64) += M_tile * X_tile   (4 ksteps x 4 d-tiles of WMMA)
// Deterministic: partials written per slice, no float atomics.
// ---------------------------------------------------------------------------
__global__ __launch_bounds__(256) void hyp_pair(const float* __restrict__ X,
                                                const float* __restrict__ x2s,
                                                const float* __restrict__ f1s,
                                                const float* __restrict__ f2s,
                                                const float* __restrict__ attb,
                                                float* __restrict__ sAPart,
                                                float* __restrict__ supPart,
                                                int n) {
  const int lane  = threadIdx.x & 31;
  const int wave  = threadIdx.x >> 5;
  const int gw    = blockIdx.x * 8 + wave;   // n/4 waves total
  const int strip = gw >> 2;
  const int slice = gw & 3;
  const int i0    = strip * 16;
  const int jtiles = n >> 6;                 // (n/16)/4 tiles per slice
  const int j0base = slice * (jtiles * 16);
  const int half  = lane >> 4;
  const int l16   = lane & 15;

  __shared__ float lds[8][16 * 17];
  float* mt = lds[wave];

  float x2i[8], f1i[8];
#pragma unroll
  for (int r = 0; r < 8; ++r) {
    const int i = i0 + r + 8 * half;
    x2i[r] = x2s[i];
    f1i[r] = f1s[i];
  }
  const float bb = attb[0];

  v8f sup0 = {}, sup1 = {}, sup2 = {}, sup3 = {};
  float sAcc[8];
#pragma unroll
  for (int r = 0; r < 8; ++r) sAcc[r] = 0.f;

  // A-frag base for GEMM1: lanes 0-15 -> K={4k,4k+1}, lanes 16-31 -> K={4k+2,4k+3}
  const float* arow = X + (i0 + l16) * 64 + 2 * half;

  for (int jt = 0; jt < jtiles; ++jt) {
    const int j0 = j0base + jt * 16;
    const float* brow = X + (j0 + l16) * 64 + 2 * half;

    // ---- GEMM1: G = X_i * X_j^T, K = 64 ----
    v8f g = {};
#pragma unroll
    for (int k = 0; k < 16; ++k) {
      v2f a = *(const v2f*)(arow + 4 * k);
      v2f b = *(const v2f*)(brow + 4 * k);
      g = WMMA_F32(a, b, g);
    }

    // ---- elementwise on the G tile ----
    const float x2j = x2s[j0 + l16];
    const float f2j = f2s[j0 + l16];
#pragma unroll
    for (int r = 0; r < 8; ++r) {
      const float xy   = g[r];
      const float Ac   = 1.f - 2.f * xy + x2j;                       // c = 1
      const float den  = fmaxf(1.f - 2.f * xy + x2i[r] * x2j, MIN_NORM);
      const float inv  = frcp(den);
      const float alpha = -Ac * inv;
      const float beta  = (1.f - x2i[r]) * inv;
      const float sn2  = alpha * alpha * x2i[r] + 2.f * alpha * beta * xy
                       + beta * beta * x2j;                          // |sub|^2
      const float snc  = fmaxf(fsqrt(fmaxf(sn2, 0.f)), MIN_NORM);
      const float at   = artanh_f(fminf(snc, 1.f - ATANH_EPS));
      const float scale = fmaxf(1.f - x2i[r], MIN_NORM) * at * frcp(snc);
      const float att  = frcp(1.f + __expf(-(f1i[r] + f2j + bb)));
      const float w    = att * scale;
      sAcc[r] += w * alpha;
      mt[(r + 8 * half) * 17 + l16] = w * beta;                      // M tile -> LDS
    }

    // ---- GEMM2: sup(16x64) += M_tile(16x16) * X[j0:j0+16, 0:64] ----
#pragma unroll
    for (int kk = 0; kk < 4; ++kk) {
      v2f a;
      a.x = mt[l16 * 17 + 4 * kk + 2 * half];
      a.y = mt[l16 * 17 + 4 * kk + 2 * half + 1];
      const int krow = j0 + 4 * kk + 2 * half;
      const float* b0 = X + krow * 64 + l16;
      v2f b;
      b.x = b0[0];  b.y = b0[64];        sup0 = WMMA_F32(a, b, sup0);
      b.x = b0[16]; b.y = b0[80];        sup1 = WMMA_F32(a, b, sup1);
      b.x = b0[32]; b.y = b0[96];        sup2 = WMMA_F32(a, b, sup2);
      b.x = b0[48]; b.y = b0[112];       sup3 = WMMA_F32(a, b, sup3);
    }
  }

  // ---- write slice partials ----
#pragma unroll
  for (int r = 0; r < 8; ++r) {
    float v = sAcc[r];
    v += __shfl_xor(v, 1, 32);
    v += __shfl_xor(v, 2, 32);
    v += __shfl_xor(v, 4, 32);
    v += __shfl_xor(v, 8, 32);           // reduce within each half (rows differ per half)
    if (l16 == 0) sAPart[slice * n + i0 + r + 8 * half] = v;
  }
#pragma unroll
  for (int r = 0; r < 8; ++r) {
    const int i = i0 + r + 8 * half;
    float* dst = supPart + (size_t)(slice * n + i) * 64 + l16;
    dst[0]  = sup0[r];
    dst[16] = sup1[r];
    dst[32] = sup2[r];
    dst[48] = sup3[r];
  }
}

// ---------------------------------------------------------------------------
// Kernel 3: reduce slice partials, then expmap(x,u) -> mobius_add -> proj.
// One wave per row; d=64 -> 2 elements per lane; full-wave shuffle reductions.
// ---------------------------------------------------------------------------
__global__ __launch_bounds__(256) void hyp_final(const float* __restrict__ X,
                                                 const float* __restrict__ x2s,
                                                 const float* __restrict__ sAPart,
                                                 const float* __restrict__ supPart,
                                                 float* __restrict__ out,
                                                 int n) {
  const int lane = threadIdx.x & 31;
  const int wave = threadIdx.x >> 5;
  const int row  = blockIdx.x * 8 + wave;

  const float sAr = sAPart[row] + sAPart[n + row] + sAPart[2 * n + row] + sAPart[3 * n + row];
  const float p0 = X[row * 64 + lane];
  const float p1 = X[row * 64 + lane + 32];
  float u0 = sAr * p0, u1 = sAr * p1;
#pragma unroll
  for (int s = 0; s < 4; ++s) {
    const float* sp = supPart + (size_t)(s * n + row) * 64;
    u0 += sp[lane];
    u1 += sp[lane + 32];
  }

  float un2 = u0 * u0 + u1 * u1;
  float pu  = p0 * u0 + p1 * u1;
#pragma unroll
  for (int m = 16; m >= 1; m >>= 1) {
    un2 += __shfl_xor(un2, m, 32);
    pu  += __shfl_xor(pu,  m, 32);
  }

  const float x2i = x2s[row];
  const float un  = fmaxf(fsqrt(un2), MIN_NORM);
  const float irn = frcp(un);
  const float lam = 2.f * frcp(fmaxf(1.f - x2i, MIN_NORM));
  const float th  = tanhf(0.5f * lam * un);      // second = th * u / un, |second| = th
  const float s2  = th * th;
  const float ps  = th * pu * irn;               // p . second
  const float iden = frcp(fmaxf(1.f + 2.f * ps + x2i * s2, MIN_NORM));
  const float cP  = (1.f + 2.f * ps + s2) * iden;
  const float cS  = (1.f - x2i) * (th * irn) * iden;

  float o0 = cP * p0 + cS * u0;
  float o1 = cP * p1 + cS * u1;
  float on2 = o0 * o0 + o1 * o1;
#pragma unroll
  for (int m = 16; m >= 1; m >>= 1) on2 += __shfl_xor(on2, m, 32);
  const float on   = fmaxf(fsqrt(on2), MIN_NORM);
  const float maxn = 1.f - PROJ_EPS;
  const float f    = (on > maxn) ? (maxn * frcp(on)) : 1.f;

  out[row * 64 + lane]      = f * o0;
  out[row * 64 + lane + 32] = f * o1;
}

// ---------------------------------------------------------------------------
extern "C" void kernel_launch(void* const* d_in, const int* in_sizes, int n_in,
                              void* d_out, int out_size, void* d_ws, size_t ws_size,
                              hipStream_t stream) {
  const float* X    = (const float*)d_in[0];   // (n, 64) f32
  const float* attw = (const float*)d_in[1];   // (1, 128) f32
  const float* attb = (const float*)d_in[2];   // (1,) f32
  // d_in[3] (adj) is unused by the reference.

  const int n = in_sizes[0] / 64;              // 1024

  float* W       = (float*)d_ws;
  float* x2s     = W;                          // n
  float* f1s     = W + n;                      // n
  float* f2s     = W + 2 * n;                  // n
  float* sAPart  = W + 3 * n;                  // 4*n
  float* supPart = W + 7 * n;                  // 4*n*64
  float* out     = (float*)d_out;

  hyp_stats<<<n / 8,  256, 0, stream>>>(X, attw, x2s, f1s, f2s);
  hyp_pair <<<n / 32, 256, 0, stream>>>(X, x2s, f1s, f2s, attb, sAPart, supPart, n);
  hyp_final<<<n / 8,  256, 0, stream>>>(X, x2s, sAPart, supPart, out, n);
}